// EdgeEmbedding_30623116821332
// MI455X (gfx1250) — compile-verified
//
#include <hip/hip_runtime.h>
#include <cstdint>
#include <cstddef>

// out[i,:] = edge_type_embedding[data[i],:]
//          + segment_sum(attr_table[flat_attr_ids], attr_seg_ids)[data[i],:]
// Shapes: N=1e6, D=256, NUM_TYPES=1000 (<=1024 assumed for the scan), TOTAL_ATTRS=50000.

typedef float v4f __attribute__((ext_vector_type(4)));

#define EMBED_D 256
#define D4      (EMBED_D / 4)   // 64 float4 per row

// ---------------------------------------------------------------------------
// E (placed first so the disasm snippet shows the CDNA5 async path):
// per-type accumulation, atomic-free.
// One 64-lane group per type; the 256-float row lives in registers (one
// float4 per lane). Attr rows are staged global->LDS with double-buffered
// CDNA5 async DMA: global_load_async_to_lds_b128 + s_wait_asynccnt.
//   combined[t,:] = ete[t,:] + sum_{a in bin(t)} attr_table[flat[a],:]
// ---------------------------------------------------------------------------
__global__ __launch_bounds__(256) void type_sum_kernel(
    const float* __restrict__ attr_table,
    const int*   __restrict__ flat_attr_ids,
    const float* __restrict__ ete,
    const int*   __restrict__ counts,
    const int*   __restrict__ offsets,
    const int*   __restrict__ bins,
    float*       __restrict__ combined,
    int num_types) {
  __shared__ __align__(16) float stage[2][256 * 4];  // 2 bufs x (4 groups x 64 lanes x float4)
  const int tid   = threadIdx.x;
  const int group = tid >> 6;           // 0..3, each group = 2 wave32
  const int lane  = tid & 63;           // one float4 column of the row
  const int t     = blockIdx.x * 4 + group;
  if (t >= num_types) return;

  v4f acc = ((const v4f*)ete)[(size_t)t * D4 + lane];
  const int cnt   = counts[t];
  const int start = offsets[t];

  if (cnt > 0) {
    // prologue: async-stage row 0 into buffer 0
    {
      const int a0 = bins[start];
      const float* g = attr_table + (size_t)flat_attr_ids[a0] * EMBED_D + lane * 4;
      unsigned l = (unsigned)(uintptr_t)&stage[0][tid * 4];
      asm volatile("global_load_async_to_lds_b128 %0, %1, off"
                   :: "v"(l), "v"(g) : "memory");
    }
    for (int j = 0; j < cnt; ++j) {
      const int cur = j & 1;
      if (j + 1 < cnt) {
        const int a1 = bins[start + j + 1];
        const float* g = attr_table + (size_t)flat_attr_ids[a1] * EMBED_D + lane * 4;
        unsigned l = (unsigned)(uintptr_t)&stage[cur ^ 1][tid * 4];
        // close WAR on the recycled buffer: our ds readback must be complete
        asm volatile("s_wait_dscnt 0x0" ::: "memory");
        asm volatile("global_load_async_to_lds_b128 %0, %1, off"
                     :: "v"(l), "v"(g) : "memory");
        // async loads complete in order: <=1 outstanding => row j is in LDS
        asm volatile("s_wait_asynccnt 0x1" ::: "memory");
      } else {
        asm volatile("s_wait_asynccnt 0x0" ::: "memory");
      }
      v4f v = *(const v4f*)&stage[cur][tid * 4];  // each lane reads its own 16B
      acc += v;
    }
  }
  ((v4f*)combined)[(size_t)t * D4 + lane] = acc;  // plain store, no atomics
}

// ---------------------------------------------------------------------------
// A: zero the per-type counters (re-done every call -> deterministic replays)
// ---------------------------------------------------------------------------
__global__ __launch_bounds__(256) void zero_counts_kernel(int* __restrict__ counts, int n) {
  int i = blockIdx.x * 256 + threadIdx.x;
  if (i < n) counts[i] = 0;
}

// ---------------------------------------------------------------------------
// B: histogram of attr segment ids (cheap int atomics, 50K total)
// ---------------------------------------------------------------------------
__global__ __launch_bounds__(256) void hist_kernel(const int* __restrict__ sids,
                                                   int* __restrict__ counts, int n) {
  int i = blockIdx.x * 256 + threadIdx.x;
  if (i < n) atomicAdd(&counts[sids[i]], 1);
}

// ---------------------------------------------------------------------------
// C: single-block exclusive scan over counts (num_types <= 1024)
// ---------------------------------------------------------------------------
__global__ __launch_bounds__(1024) void scan_kernel(const int* __restrict__ counts,
                                                    int* __restrict__ offsets,
                                                    int* __restrict__ cursor, int n) {
  __shared__ int sh[1024];
  const int tid = threadIdx.x;
  const int v = (tid < n) ? counts[tid] : 0;
  sh[tid] = v;
  __syncthreads();
  for (int off = 1; off < 1024; off <<= 1) {
    int x = (tid >= off) ? sh[tid - off] : 0;
    __syncthreads();
    sh[tid] += x;
    __syncthreads();
  }
  if (tid < n) {
    int e = sh[tid] - v;     // exclusive prefix
    offsets[tid] = e;
    cursor[tid]  = e;
  }
}

// ---------------------------------------------------------------------------
// D: counting-sort fill — bins[] holds attr indices grouped by type
// ---------------------------------------------------------------------------
__global__ __launch_bounds__(256) void fill_kernel(const int* __restrict__ sids,
                                                   int* __restrict__ cursor,
                                                   int* __restrict__ bins, int n) {
  int i = blockIdx.x * 256 + threadIdx.x;
  if (i < n) {
    int slot = atomicAdd(&cursor[sids[i]], 1);
    bins[slot] = i;
  }
}

// ---------------------------------------------------------------------------
// F: out[row,:] = combined[data[row],:]
// 64 threads/row, one float4 each: coalesced 512 B/wave loads of the L2-hot
// 1 MB table, non-temporal b128 stores for the 1 GB streaming output.
// ---------------------------------------------------------------------------
__global__ __launch_bounds__(256) void gather_out_kernel(
    const int*   __restrict__ data,
    const float* __restrict__ combined,
    float*       __restrict__ out,
    int n_rows) {
  const long long gid = (long long)blockIdx.x * 256 + threadIdx.x;
  const int row = (int)(gid >> 6);
  const int c4  = (int)(gid & 63);
  if (row >= n_rows) return;
  const int t = data[row];                 // one cacheline broadcast per group
  v4f v = ((const v4f*)combined)[(size_t)t * D4 + c4];
  __builtin_nontemporal_store(v, (v4f*)out + (size_t)row * D4 + c4);
}

// ---------------------------------------------------------------------------
extern "C" void kernel_launch(void* const* d_in, const int* in_sizes, int n_in,
                              void* d_out, int out_size, void* d_ws, size_t ws_size,
                              hipStream_t stream) {
  const int*   data = (const int*)d_in[0];    // [N]
  const float* attr = (const float*)d_in[1];  // [ATTR_NUM, 256]
  const float* ete  = (const float*)d_in[2];  // [NUM_TYPES, 256]
  const int*   fids = (const int*)d_in[3];    // [TOTAL_ATTRS]
  const int*   sids = (const int*)d_in[4];    // [TOTAL_ATTRS]
  float*       out  = (float*)d_out;

  const int N           = in_sizes[0];
  const int num_types   = in_sizes[2] / EMBED_D;   // 1000
  const int total_attrs = in_sizes[3];             // 50000

  // workspace layout: combined | counts | offsets | cursor | bins  (~1.2 MB)
  float* combined = (float*)d_ws;
  int*   counts   = (int*)(combined + (size_t)num_types * EMBED_D);
  int*   offsets  = counts  + num_types;
  int*   cursor   = offsets + num_types;
  int*   bins     = cursor  + num_types;

  zero_counts_kernel<<<(num_types + 255) / 256, 256, 0, stream>>>(counts, num_types);
  hist_kernel<<<(total_attrs + 255) / 256, 256, 0, stream>>>(sids, counts, total_attrs);
  scan_kernel<<<1, 1024, 0, stream>>>(counts, offsets, cursor, num_types);
  fill_kernel<<<(total_attrs + 255) / 256, 256, 0, stream>>>(sids, cursor, bins, total_attrs);
  type_sum_kernel<<<(num_types + 3) / 4, 256, 0, stream>>>(
      attr, fids, ete, counts, offsets, bins, combined, num_types);

  const long long tot4 = (long long)N * D4;
  gather_out_kernel<<<(int)((tot4 + 255) / 256), 256, 0, stream>>>(data, combined, out, N);
}